// Model_38113539785434
// MI455X (gfx1250) — compile-verified
//
#include <hip/hip_runtime.h>
#include <hip/hip_bf16.h>
#include <stdint.h>

// Bilateral filter 9x9, sigma_s=3, sigma_r=0.1, reflect padding, 4096x4096 f32.
// TDM (tensor_load_to_lds) fills the LDS tile for interior blocks; border
// blocks use per-lane loads with reflect indexing. One v_exp_f32 per tap
// (the compiler folds __expf's log2e into the fmaf constants).

#define H 4096
#define W 4096
#define R 4               // kernel radius
#define TILE 64           // output tile per block
#define LW (TILE + 2 * R) // 72: LDS tile width (with halo)
#define NTHREADS 256

typedef unsigned int u32x4 __attribute__((ext_vector_type(4)));
typedef int          i32x8 __attribute__((ext_vector_type(8)));
typedef int          i32x4 __attribute__((ext_vector_type(4)));

__device__ __forceinline__ int reflect_idx(int i, int n) {
    // jnp.pad mode='reflect': -1 -> 1, n -> n-2 (single reflection, r << n)
    i = (i < 0) ? -i : i;
    i = (i >= n) ? (2 * n - 2 - i) : i;
    return i;
}

__global__ __launch_bounds__(NTHREADS)
void bilateral_kernel(const float* __restrict__ img, float* __restrict__ out) {
    __shared__ float tile[LW * LW];

    const int tid = threadIdx.x;
    const int bx  = blockIdx.x;
    const int by  = blockIdx.y;

    const int gx0 = bx * TILE - R; // global coord of tile[0][0]
    const int gy0 = by * TILE - R;

    const bool interior = (gx0 >= 0) && (gy0 >= 0) &&
                          (gx0 + LW <= W) && (gy0 + LW <= H);

    if (interior) {
        // ---- TDM path: one DMA of the 72x72 tile (strided 2D) into LDS ----
        if (tid < 32) { // wave 0 only issues the tensor op
            uint32_t lds_base = (uint32_t)(uintptr_t)(&tile[0]); // LDS byte offset
            uint64_t ga = (uint64_t)(uintptr_t)img +
                          (uint64_t)((int64_t)gy0 * W + gx0) * 4u;

            // D# group 0 (128b): count=1 | lds_addr | global_addr[56:0] | type=2
            u32x4 g0;
            g0.x = 1u;                                   // count=1, user mode
            g0.y = lds_base;                             // lds_addr (bytes)
            g0.z = (uint32_t)ga;                         // global_addr[31:0]
            g0.w = ((uint32_t)(ga >> 32) & 0x01FFFFFFu)  // global_addr[56:32]
                   | (2u << 30);                         // type = 2 ("image")

            // D# group 1 (256b)
            i32x8 g1;
            g1[0] = (int)(2u << 16);            // wg_mask=0, data_size=2 (4B)
            g1[1] = (int)((uint32_t)W << 16);   // tensor_dim0[15:0] in hi16
            g1[2] = (int)(((uint32_t)W >> 16) | ((uint32_t)H << 16)); // dim0 hi | dim1 lo
            g1[3] = (int)(((uint32_t)H >> 16) | ((uint32_t)LW << 16)); // dim1 hi | tile_dim0
            g1[4] = (int)((uint32_t)LW);        // tile_dim1=72, tile_dim2=0
            g1[5] = (int)((uint32_t)W);         // tensor_dim0_stride[31:0] = 4096
            g1[6] = 0;                          // stride0[47:32] | stride1[15:0]
            g1[7] = 0;                          // stride1[47:16]

            i32x4 gz4 = {0, 0, 0, 0};           // groups 2/3 unused (2D tensor)
            i32x8 gz8 = {0, 0, 0, 0, 0, 0, 0, 0};

            __builtin_amdgcn_tensor_load_to_lds(g0, g1, gz4, gz4, gz8, 0);
            __builtin_amdgcn_s_wait_tensorcnt(0);
        }
    } else {
        // ---- Border path: per-lane loads with reflect index mapping ----
        for (int i = tid; i < LW * LW; i += NTHREADS) {
            int ly = i / LW;
            int lx = i - ly * LW;
            int gy = reflect_idx(gy0 + ly, H);
            int gx = reflect_idx(gx0 + lx, W);
            tile[i] = img[gy * W + gx];
        }
    }
    __syncthreads();

    // ---- Compute: 256 threads as 64 wide x 4 rows; each does 16 rows ----
    const int lx    = tid & 63;        // consecutive lanes -> consecutive LDS dwords
    const int ybase = (tid >> 6) * 16; // 4 row-groups of 16

    const float inv_range = 50.0f;          // 1 / (2 * 0.1^2)
    const float inv_spat  = 1.0f / 18.0f;   // 1 / (2 * 3^2)

    for (int k = 0; k < 16; ++k) {
        const int ly = ybase + k;
        const float c = tile[(ly + R) * LW + (lx + R)];

        float num = 0.0f;
        float den = 0.0f;
#pragma unroll
        for (int dy = 0; dy < 9; ++dy) {
            const int row = (ly + dy) * LW + lx;
#pragma unroll
            for (int dx = 0; dx < 9; ++dx) {
                const float v = tile[row + dx];
                const float d = v - c;
                // constant spatial term folded into the single exp per tap
                const float sp = (float)((dx - R) * (dx - R) + (dy - R) * (dy - R)) * inv_spat;
                const float w  = __expf(fmaf(-(d * d), inv_range, -sp));
                num = fmaf(w, v, num);
                den += w;
            }
        }

        const int gx = bx * TILE + lx;
        const int gy = by * TILE + ly;
        out[gy * W + gx] = num / den;
    }
}

extern "C" void kernel_launch(void* const* d_in, const int* in_sizes, int n_in,
                              void* d_out, int out_size, void* d_ws, size_t ws_size,
                              hipStream_t stream) {
    (void)in_sizes; (void)n_in; (void)out_size; (void)d_ws; (void)ws_size;
    const float* img = (const float*)d_in[0];
    float* out = (float*)d_out;

    dim3 grid(W / TILE, H / TILE);
    dim3 block(NTHREADS);
    bilateral_kernel<<<grid, block, 0, stream>>>(img, out);
}